// Quantizer_609885356393
// MI455X (gfx1250) — compile-verified
//
#include <hip/hip_runtime.h>
#include <hip/hip_bf16.h>

typedef float v2f __attribute__((ext_vector_type(2)));
typedef float v4f __attribute__((ext_vector_type(4)));
typedef float v8f __attribute__((ext_vector_type(8)));

#define K_CODES 512
#define DDIM    256
#define NROWS   131072
#define BATCH   32.0f
#define DECAY_F 0.99f
#define OMD_F   0.01f
#define EPSV    1e-5f
#define COMMIT  0.25f

// d_out offsets (floats), tuple order: ste, loss, enc, codebook_new, n_new, e_new
#define OUT_STE  0LL
#define OUT_LOSS 33554432LL
#define OUT_ENC  33554433LL
#define OUT_CB   33685505LL
#define OUT_N    33816577LL
#define OUT_E    33817089LL

// d_ws offsets (floats)
#define WS_COUNTS 0
#define WS_UPD    512
#define WS_LOSS   131584
#define WS_CNORM  131585

__global__ __launch_bounds__(256) void vq_prep(const float* __restrict__ cb,
                                               float* __restrict__ ws) {
  int k = blockIdx.x;         // 0..511 (one code per block)
  int t = threadIdx.x;        // 0..255 (one dim per thread)
  int gid = k * 256 + t;
  ws[WS_UPD + gid] = 0.0f;               // zero code_update accumulator
  if (gid < K_CODES) ws[WS_COUNTS + gid] = 0.0f;
  if (gid == 0) ws[WS_LOSS] = 0.0f;
  float v = cb[k * DDIM + t];
  float sq = v * v;
  #pragma unroll
  for (int m = 16; m >= 1; m >>= 1) sq += __shfl_xor(sq, m, 32);
  __shared__ float s[8];
  if ((t & 31) == 0) s[t >> 5] = sq;
  __syncthreads();
  if (t == 0) {
    float tot = 0.0f;
    #pragma unroll
    for (int i = 0; i < 8; ++i) tot += s[i];
    ws[WS_CNORM + k] = tot;              // ||c_k||^2
  }
}

__global__ __launch_bounds__(256) void vq_main(const float* __restrict__ z,
                                               const float* __restrict__ cb,
                                               float* __restrict__ out,
                                               float* __restrict__ ws) {
  constexpr int APAD = 260;              // padded row stride (floats) for LDS banking
  __shared__ float sA[32 * APAD];        // 32 rows x 256 of z, padded (~33 KB)
  __shared__ float sBestV[8][32];
  __shared__ int   sBestI[8][32];
  __shared__ int   sEnc[32];
  __shared__ float sLoss[8];

  const int t = threadIdx.x;
  const int lane = t & 31;
  const int wave = t >> 5;               // 8 waves; wave w owns codes [w*64, w*64+64)
  const int row_base = blockIdx.x * 32;

  // ---- stage 32 rows of flat z into LDS (coalesced float4, non-temporal:
  //      z is streamed once; keep the hot codebook resident in L2) ----
  {
    const v4f* zsrc = reinterpret_cast<const v4f*>(z + (size_t)row_base * DDIM);
    #pragma unroll
    for (int i = 0; i < 8; ++i) {
      int f4 = i * 256 + t;              // 0..2047 float4s
      int e  = f4 * 4;
      int r  = e >> 8;                   // row 0..31
      int c  = e & 255;                  // col, multiple of 4
      *reinterpret_cast<v4f*>(&sA[r * APAD + c]) =
          __builtin_nontemporal_load(&zsrc[f4]);
    }
  }

  // ---- per-lane WMMA operand addressing ----
  // A (16x4 f32): lane L -> M = L&15, holds K = {kk, kk+1} with kk = 2*(L>>4)
  // B (4x16 f32): lane L -> N = L&15, holds K = {kk, kk+1};  B[k][n] = codebook[n][k]
  const int n  = lane & 15;
  const int kk = (lane >> 4) * 2;
  int code_idx[4];
  float cn[4];
  const float* bptr[4];
  #pragma unroll
  for (int i = 0; i < 4; ++i) {
    code_idx[i] = wave * 64 + i * 16 + n;
    cn[i] = ws[WS_CNORM + code_idx[i]];
    bptr[i] = cb + (size_t)code_idx[i] * DDIM + kk;
  }

  __syncthreads();

  v8f acc[2][4];
  const v8f vzero = {0.f,0.f,0.f,0.f,0.f,0.f,0.f,0.f};
  #pragma unroll
  for (int rt = 0; rt < 2; ++rt)
    #pragma unroll
    for (int i = 0; i < 4; ++i) acc[rt][i] = vzero;

  const int a_base0 = n * APAD + kk;            // M-tile 0 (rows 0..15)
  const int a_base1 = (16 + n) * APAD + kk;     // M-tile 1 (rows 16..31)

  for (int k = 0; k < DDIM; k += 4) {
    v2f a0 = *reinterpret_cast<const v2f*>(&sA[a_base0 + k]);
    v2f a1 = *reinterpret_cast<const v2f*>(&sA[a_base1 + k]);
    #pragma unroll
    for (int i = 0; i < 4; ++i) {
      v2f b = *reinterpret_cast<const v2f*>(bptr[i] + k);
      acc[0][i] = __builtin_amdgcn_wmma_f32_16x16x4_f32(
          false, a0, false, b, (short)0, acc[0][i], false, false);
      acc[1][i] = __builtin_amdgcn_wmma_f32_16x16x4_f32(
          false, a1, false, b, (short)0, acc[1][i], false, false);
    }
  }

  // ---- argmin of (||c||^2 - 2*S) per row ----
  // C/D layout: lane<16 -> N=lane, M=vgpr j; lane>=16 -> N=lane-16, M=j+8
  float bv[2][8];
  int   bi[2][8];
  #pragma unroll
  for (int rt = 0; rt < 2; ++rt) {
    #pragma unroll
    for (int j = 0; j < 8; ++j) {
      float best = cn[0] - 2.0f * acc[rt][0][j];
      int   bidx = code_idx[0];
      #pragma unroll
      for (int i = 1; i < 4; ++i) {
        float v = cn[i] - 2.0f * acc[rt][i][j];
        if (v < best || (v == best && code_idx[i] < bidx)) { best = v; bidx = code_idx[i]; }
      }
      // butterfly across the 16 lanes of each half (stays within a half for masks<16)
      #pragma unroll
      for (int m = 8; m >= 1; m >>= 1) {
        float ov = __shfl_xor(best, m, 32);
        int   oi = __shfl_xor(bidx, m, 32);
        if (ov < best || (ov == best && oi < bidx)) { best = ov; bidx = oi; }
      }
      bv[rt][j] = best; bi[rt][j] = bidx;
    }
  }

  if (n == 0) {                          // lane 0 (rows j) and lane 16 (rows j+8)
    int half = lane >> 4;
    #pragma unroll
    for (int rt = 0; rt < 2; ++rt)
      #pragma unroll
      for (int j = 0; j < 8; ++j) {
        int row = rt * 16 + half * 8 + j;
        sBestV[wave][row] = bv[rt][j];
        sBestI[wave][row] = bi[rt][j];
      }
  }
  __syncthreads();

  // cross-wave reduction: one thread per row
  if (t < 32) {
    float best = sBestV[0][t]; int bidx = sBestI[0][t];
    #pragma unroll
    for (int w = 1; w < 8; ++w) {
      float v = sBestV[w][t]; int i2 = sBestI[w][t];
      if (v < best || (v == best && i2 < bidx)) { best = v; bidx = i2; }
    }
    sEnc[t] = bidx;
    out[OUT_ENC + row_base + t] = (float)bidx;       // encoding (harness: float out)
    atomicAdd(&ws[WS_COUNTS + bidx], 1.0f);          // counts segment-sum
  }
  __syncthreads();

  // ---- outputs: code_ste (NT store, write-once stream), loss, code_update ----
  float lsum = 0.0f;
  for (int r = 0; r < 32; ++r) {
    int enc = sEnc[r];
    float zz = sA[r * APAD + t];
    float c  = cb[(size_t)enc * DDIM + t];           // gather code row (coalesced, RT)
    __builtin_nontemporal_store(zz + (c - zz),
        &out[OUT_STE + (size_t)(row_base + r) * DDIM + t]);  // STE
    float d = zz - c;
    lsum += d * d;
    atomicAdd(&ws[WS_UPD + enc * DDIM + t], zz);     // idx.T @ flat segment-sum
  }
  #pragma unroll
  for (int m = 16; m >= 1; m >>= 1) lsum += __shfl_xor(lsum, m, 32);
  if (lane == 0) sLoss[wave] = lsum;
  __syncthreads();
  if (t == 0) {
    float tot = 0.0f;
    #pragma unroll
    for (int i = 0; i < 8; ++i) tot += sLoss[i];
    atomicAdd(&ws[WS_LOSS], tot);
  }
}

__global__ __launch_bounds__(256) void vq_final(const float* __restrict__ n_i,
                                                const float* __restrict__ e_i,
                                                float* __restrict__ out,
                                                const float* __restrict__ ws) {
  int k = blockIdx.x;                    // code
  int d = threadIdx.x;                   // dim
  float counts = ws[WS_COUNTS + k];
  float nn = DECAY_F * n_i[k] + OMD_F * counts;
  nn = (nn + EPSV) / (BATCH + (float)K_CODES * EPSV) * BATCH;   // Laplace smoothing
  float en = DECAY_F * e_i[(size_t)k * DDIM + d] + OMD_F * ws[WS_UPD + k * DDIM + d];
  out[OUT_E  + (size_t)k * DDIM + d] = en;
  out[OUT_CB + (size_t)k * DDIM + d] = en / nn;
  if (d == 0) out[OUT_N + k] = nn;
  if (k == 0 && d == 0) out[OUT_LOSS] = COMMIT * ws[WS_LOSS] / 33554432.0f;
}

extern "C" void kernel_launch(void* const* d_in, const int* in_sizes, int n_in,
                              void* d_out, int out_size, void* d_ws, size_t ws_size,
                              hipStream_t stream) {
  (void)in_sizes; (void)n_in; (void)out_size; (void)ws_size;
  const float* z  = (const float*)d_in[0];
  const float* cb = (const float*)d_in[1];
  const float* ni = (const float*)d_in[2];
  const float* ei = (const float*)d_in[3];
  float* out = (float*)d_out;
  float* ws  = (float*)d_ws;

  vq_prep <<<K_CODES, 256, 0, stream>>>(cb, ws);
  vq_main <<<NROWS / 32, 256, 0, stream>>>(z, cb, out, ws);
  vq_final<<<K_CODES, 256, 0, stream>>>(ni, ei, out, ws);
}